// Sg2ImModel_22935125360838
// MI455X (gfx1250) — compile-verified
//
#include <hip/hip_runtime.h>
#include <hip/hip_bf16.h>

#define EMB   64
#define GDIM  128
#define HID   512
#define NOBJ  64
#define NTRI  192
#define NIMG  8
#define IMGHW 64
#define NTOT  1152   // 2*HID + GDIM
#define KMAX  512

typedef __attribute__((ext_vector_type(2))) float v2f;
typedef __attribute__((ext_vector_type(8))) float v8f;

// ---------------------------------------------------------------------------
// WMMA f32 GEMM with LDS-staged A tile:
//   C[M,N] = relu(A[M,K] @ W[K,N] + bias[N])
// Block = 256 threads (8 wave32), covers a 16-row x 128-col output stripe.
// The 16xK A tile is loaded to LDS once (coalesced) and shared by all 8
// waves; each wave computes one 16x16 tile via V_WMMA_F32_16X16X4_F32.
// LDS row stride is K+2 floats: all K here are multiples of 64, so an
// unpadded stride would land all 16 lanes on one bank.
// Fragment layouts per CDNA5 ISA:
//   A: lanes 0-15 rows M=0..15 hold {K,K+1}; lanes 16-31 hold {K+2,K+3}
//   B: lanes 0-15 cols N=0..15 hold {K,K+1}; lanes 16-31 hold {K+2,K+3}
//   C/D: VGPR r -> row r (lanes 0-15) / row r+8 (lanes 16-31)
// Requires M%16==0, N%128==0, K%4==0, K<=KMAX. EXEC all-ones per wave.
// ---------------------------------------------------------------------------
__global__ void wmma_gemm_bias_relu(const float* __restrict__ A,
                                    const float* __restrict__ W,
                                    const float* __restrict__ bias,
                                    float* __restrict__ C,
                                    int M, int K, int N) {
  __shared__ float As[16 * (KMAX + 2)];
  const int LDA = K + 2;

  int nblk = N >> 7;                    // 128-col stripes
  int mt   = (int)blockIdx.x / nblk;
  int nb   = (int)blockIdx.x % nblk;

  int tid  = (int)threadIdx.x;
  int wave = tid >> 5;                  // 0..7 -> which 16-col tile
  int lane = tid & 31;
  int half = lane >> 4;                 // 0: {K,K+1}  1: {K+2,K+3}
  int l    = lane & 15;

  // Cooperative, coalesced fill of the 16xK A tile.
  int total = 16 * K;
  for (int i = tid; i < total; i += 256) {
    int r = i / K;
    int c = i - r * K;
    As[r * LDA + c] = A[(size_t)(mt * 16 + r) * (size_t)K + c];
  }
  __syncthreads();

  int col = (nb * 8 + wave) * 16 + l;   // this lane's output column
  const float* Wcol = W + col;
  const float* Alds = As + l * LDA;     // row l of the LDS tile

  v8f acc = (v8f)0.0f;
  for (int k = 0; k < K; k += 4) {
    int ka = k + half * 2;
    v2f a, b;
    a.x = Alds[ka];                     // ds_load_b64 (8B aligned: K even)
    a.y = Alds[ka + 1];
    b.x = Wcol[(size_t)ka * (size_t)N];
    b.y = Wcol[(size_t)(ka + 1) * (size_t)N];
    acc = __builtin_amdgcn_wmma_f32_16x16x4_f32(false, a, false, b,
                                                (short)0, acc, false, false);
  }

  float bb = bias[col];
#pragma unroll
  for (int r = 0; r < 8; ++r) {
    int row = mt * 16 + half * 8 + r;
    float v = acc[r] + bb;
    C[(size_t)row * (size_t)N + col] = fmaxf(v, 0.0f);
  }
}

// ---------------------------------------------------------------------------
// Embedding gathers
// ---------------------------------------------------------------------------
__global__ void embed_obj_kernel(const float* __restrict__ emb,
                                 const int* __restrict__ objs,
                                 float* __restrict__ out) {
  int idx = blockIdx.x * blockDim.x + threadIdx.x;
  if (idx >= NOBJ * EMB) return;
  int i = idx >> 6, e = idx & 63;
  out[i * EMB + e] = emb[objs[i] * EMB + e];
}

__global__ void embed_pred_kernel(const float* __restrict__ emb,
                                  const int* __restrict__ triples,
                                  float* __restrict__ out) {
  int idx = blockIdx.x * blockDim.x + threadIdx.x;
  if (idx >= NTRI * EMB) return;
  int t = idx >> 6, e = idx & 63;
  out[t * EMB + e] = emb[triples[t * 3 + 1] * EMB + e];
}

// t = concat(obj_vecs[s], pred_vecs, obj_vecs[o])  -> (NTRI, 3*din)
__global__ void gather_t_kernel(const float* __restrict__ obj_vecs,
                                const float* __restrict__ pred_vecs,
                                const int* __restrict__ triples,
                                float* __restrict__ tmat, int din) {
  int total = NTRI * 3 * din;
  int idx = blockIdx.x * blockDim.x + threadIdx.x;
  if (idx >= total) return;
  int t = idx / (3 * din);
  int c = idx - t * 3 * din;
  int s = triples[t * 3 + 0];
  int o = triples[t * 3 + 2];
  float v;
  if (c < din)            v = obj_vecs[s * din + c];
  else if (c < 2 * din)   v = pred_vecs[t * din + (c - din)];
  else                    v = obj_vecs[o * din + (c - 2 * din)];
  tmat[idx] = v;
}

// Deterministic gather version of scatter-mean pooling.
// nt layout: new_s = nt[:, :512], new_p = nt[:, 512:640], new_o = nt[:, 640:1152]
__global__ void pool_kernel(const float* __restrict__ nt,
                            const int* __restrict__ triples,
                            float* __restrict__ pooled) {
  int idx = blockIdx.x * blockDim.x + threadIdx.x;
  if (idx >= NOBJ * HID) return;
  int i = idx >> 9;
  int d = idx & 511;
  float sum = 0.0f, cnt = 0.0f;
  for (int t = 0; t < NTRI; ++t) {
    int s = triples[t * 3 + 0];
    int o = triples[t * 3 + 2];
    if (s == i) { sum += nt[(size_t)t * NTOT + d];        cnt += 1.0f; }
    if (o == i) { sum += nt[(size_t)t * NTOT + 640 + d];  cnt += 1.0f; }
  }
  pooled[idx] = sum / fmaxf(cnt, 1.0f);
}

__global__ void copy_newp_kernel(const float* __restrict__ nt,
                                 float* __restrict__ pred_vecs) {
  int idx = blockIdx.x * blockDim.x + threadIdx.x;
  if (idx >= NTRI * GDIM) return;
  int t = idx >> 7, d = idx & 127;
  pred_vecs[t * GDIM + d] = nt[(size_t)t * NTOT + HID + d];
}

// Final box head: out[i,j] = relu(bh[i,:] @ w2[:,j] + b2[j]),  (64,512)x(512,4)
__global__ void box2_kernel(const float* __restrict__ bh,
                            const float* __restrict__ w2,
                            const float* __restrict__ b2,
                            float* __restrict__ out) {
  int idx = blockIdx.x * blockDim.x + threadIdx.x;
  if (idx >= NOBJ * 4) return;
  int i = idx >> 2, j = idx & 3;
  float acc = b2[j];
  for (int k = 0; k < HID; ++k) acc += bh[i * HID + k] * w2[k * 4 + j];
  out[idx] = fmaxf(acc, 0.0f);
}

// Bilinear box weight maps (exact reference math), wmap[o, y, x]
__global__ void wmap_kernel(const float* __restrict__ boxes,
                            float* __restrict__ wmap) {
  int idx = blockIdx.x * blockDim.x + threadIdx.x;
  if (idx >= NOBJ * IMGHW * IMGHW) return;
  int i = idx >> 12;
  int y = (idx >> 6) & 63;
  int x = idx & 63;
  float x0 = boxes[i * 4 + 0], y0 = boxes[i * 4 + 1];
  float x1 = boxes[i * 4 + 2], y1 = boxes[i * 4 + 3];
  float X = (float)x / 63.0f;
  float Y = (float)y / 63.0f;
  float gx = (X - x0) / (x1 - x0) * 2.0f - 1.0f;
  float gy = (Y - y0) / (y1 - y0) * 2.0f - 1.0f;
  float ix = ((gx + 1.0f) * 8.0f - 1.0f) * 0.5f;
  float iy = ((gy + 1.0f) * 8.0f - 1.0f) * 0.5f;
  float t0, f, lo, hi, wx, wy;
  t0 = floorf(iy); f = iy - t0;
  lo = (t0 >= 0.0f && t0 <= 7.0f) ? 1.0f : 0.0f;
  hi = (t0 + 1.0f >= 0.0f && t0 + 1.0f <= 7.0f) ? 1.0f : 0.0f;
  wy = (1.0f - f) * lo + f * hi;
  t0 = floorf(ix); f = ix - t0;
  lo = (t0 >= 0.0f && t0 <= 7.0f) ? 1.0f : 0.0f;
  hi = (t0 + 1.0f >= 0.0f && t0 + 1.0f <= 7.0f) ? 1.0f : 0.0f;
  wx = (1.0f - f) * lo + f * hi;
  wmap[idx] = wy * wx;
}

// layout[img, c, y, x] = sum_o [obj_to_img[o]==img] * obj_vecs[o,c] * wmap[o,y,x]
// Gather form: fully deterministic, no zero-init or atomics needed.
__global__ void layout_kernel(const float* __restrict__ obj_vecs,
                              const float* __restrict__ wmap,
                              const int* __restrict__ obj_to_img,
                              float* __restrict__ out) {
  int idx = blockIdx.x * blockDim.x + threadIdx.x;
  if (idx >= NIMG * GDIM * IMGHW * IMGHW) return;
  int img = idx >> 19;          // / (128*4096)
  int c   = (idx >> 12) & 127;
  int pix = idx & 4095;
  float sum = 0.0f;
  for (int o = 0; o < NOBJ; ++o) {
    if (obj_to_img[o] == img)
      sum += obj_vecs[o * GDIM + c] * wmap[o * 4096 + pix];
  }
  out[((size_t)(img * 192 + c) << 12) + pix] = sum;
}

// 3x3 SAME conv, 3 -> 64 channels, + bias + relu, into output channels 128..191
__global__ void conv_kernel(const float* __restrict__ img,
                            const float* __restrict__ w,
                            const float* __restrict__ b,
                            float* __restrict__ out) {
  int idx = blockIdx.x * blockDim.x + threadIdx.x;
  if (idx >= NIMG * 64 * IMGHW * IMGHW) return;
  int n  = idx >> 18;
  int oc = (idx >> 12) & 63;
  int y  = (idx >> 6) & 63;
  int x  = idx & 63;
  float acc = b[oc];
  for (int ic = 0; ic < 3; ++ic) {
    for (int ky = 0; ky < 3; ++ky) {
      int yy = y + ky - 1;
      if (yy < 0 || yy > 63) continue;
      for (int kx = 0; kx < 3; ++kx) {
        int xx = x + kx - 1;
        if (xx < 0 || xx > 63) continue;
        acc += img[(((size_t)n * 3 + ic) * 64 + yy) * 64 + xx] *
               w[((oc * 3 + ic) * 3 + ky) * 3 + kx];
      }
    }
  }
  out[((size_t)(n * 192 + 128 + oc) << 12) + (y << 6) + x] = fmaxf(acc, 0.0f);
}

// ---------------------------------------------------------------------------
static inline dim3 g1d(int n) { return dim3((unsigned)((n + 255) / 256)); }

// grid for the LDS-staged GEMM: (M/16) * (N/128) blocks of 256 threads
static inline dim3 ggemm(int M, int N) { return dim3((unsigned)((M / 16) * (N / 128))); }

extern "C" void kernel_launch(void* const* d_in, const int* in_sizes, int n_in,
                              void* d_out, int out_size, void* d_ws, size_t ws_size,
                              hipStream_t stream) {
  (void)in_sizes; (void)n_in; (void)out_size; (void)ws_size;

  const float* img        = (const float*)d_in[0];
  const int*   objs       = (const int*)  d_in[1];
  const int*   triples    = (const int*)  d_in[2];
  const int*   obj_to_img = (const int*)  d_in[3];
  const float* boxes_gt   = (const float*)d_in[4];
  const float* obj_emb    = (const float*)d_in[5];
  const float* pred_emb   = (const float*)d_in[6];
  // gconv layer L params at d_in[7 + L*8 + {0..7}] = w1,b1,w2,b2,w3,b3,w4,b4
  const float* box_w1 = (const float*)d_in[47];
  const float* box_b1 = (const float*)d_in[48];
  const float* box_w2 = (const float*)d_in[49];
  const float* box_b2 = (const float*)d_in[50];
  const float* conv_w = (const float*)d_in[51];
  const float* conv_b = (const float*)d_in[52];

  float* out = (float*)d_out;
  float* ws  = (float*)d_ws;

  // workspace layout (floats)
  float* obj_vecs  = ws + 0;        // 64*128
  float* pred_vecs = ws + 8192;     // 192*128
  float* tmat      = ws + 32768;    // 192*384
  float* hbuf      = ws + 106496;   // 192*512
  float* ntbuf     = ws + 204800;   // 192*1152
  float* pooled    = ws + 425984;   // 64*512
  float* h2        = ws + 458752;   // 64*512
  float* wmap      = ws + 491520;   // 64*64*64
  // total: 753,664 floats (~2.9 MB)

  // 1. embeddings
  embed_obj_kernel <<<g1d(NOBJ * EMB), 256, 0, stream>>>(obj_emb,  objs,    obj_vecs);
  embed_pred_kernel<<<g1d(NTRI * EMB), 256, 0, stream>>>(pred_emb, triples, pred_vecs);

  // 2. graph conv layers
  for (int L = 0; L < 5; ++L) {
    int din = (L == 0) ? EMB : GDIM;
    const float* w1 = (const float*)d_in[7 + L * 8 + 0];
    const float* b1 = (const float*)d_in[7 + L * 8 + 1];
    const float* w2 = (const float*)d_in[7 + L * 8 + 2];
    const float* b2 = (const float*)d_in[7 + L * 8 + 3];
    const float* w3 = (const float*)d_in[7 + L * 8 + 4];
    const float* b3 = (const float*)d_in[7 + L * 8 + 5];
    const float* w4 = (const float*)d_in[7 + L * 8 + 6];
    const float* b4 = (const float*)d_in[7 + L * 8 + 7];

    gather_t_kernel<<<g1d(NTRI * 3 * din), 256, 0, stream>>>(obj_vecs, pred_vecs,
                                                             triples, tmat, din);
    // (192, 3*din) @ (3*din, 512)
    wmma_gemm_bias_relu<<<ggemm(NTRI, HID), 256, 0, stream>>>(
        tmat, w1, b1, hbuf, NTRI, 3 * din, HID);
    // (192, 512) @ (512, 1152)
    wmma_gemm_bias_relu<<<ggemm(NTRI, NTOT), 256, 0, stream>>>(
        hbuf, w2, b2, ntbuf, NTRI, HID, NTOT);

    pool_kernel     <<<g1d(NOBJ * HID),  256, 0, stream>>>(ntbuf, triples, pooled);
    copy_newp_kernel<<<g1d(NTRI * GDIM), 256, 0, stream>>>(ntbuf, pred_vecs);

    // (64, 512) @ (512, 512)
    wmma_gemm_bias_relu<<<ggemm(NOBJ, HID), 256, 0, stream>>>(
        pooled, w3, b3, h2, NOBJ, HID, HID);
    // (64, 512) @ (512, 128) -> new obj_vecs
    wmma_gemm_bias_relu<<<ggemm(NOBJ, GDIM), 256, 0, stream>>>(
        h2, w4, b4, obj_vecs, NOBJ, HID, GDIM);
  }

  // 3. box MLP: (64,128)@(128,512) relu, then (64,512)@(512,4) relu
  wmma_gemm_bias_relu<<<ggemm(NOBJ, HID), 256, 0, stream>>>(
      obj_vecs, box_w1, box_b1, h2, NOBJ, GDIM, HID);
  box2_kernel<<<g1d(NOBJ * 4), 256, 0, stream>>>(h2, box_w2, box_b2,
                                                 out + (size_t)NIMG * 192 * 4096);

  // 4. box weight maps + layout (gather, channels 0..127)
  wmap_kernel  <<<g1d(NOBJ * 4096), 256, 0, stream>>>(boxes_gt, wmap);
  layout_kernel<<<g1d(NIMG * GDIM * 4096), 256, 0, stream>>>(obj_vecs, wmap,
                                                             obj_to_img, out);

  // 5. conv features (channels 128..191)
  conv_kernel<<<g1d(NIMG * 64 * 4096), 256, 0, stream>>>(img, conv_w, conv_b, out);
}